// LinkPredictor_27556510171654
// MI455X (gfx1250) — compile-verified
//
#include <hip/hip_runtime.h>

// ---------------------------------------------------------------------------
// GraphSAGE (2 layers, mean aggregation) for MI455X / gfx1250.
//   h1  = relu( (scatter_mean(x,  src->dst)) @ W1l^T + b1 + x  @ W1r^T )
//   out =       (scatter_mean(h1, src->dst)) @ W2l^T + b2 + h1 @ W2r^T
// N=100000 nodes, E=1250000 edges, F=64.
// Working set (~60MB) is L2-resident (192MB); scatter-add atomics dominate,
// GEMMs are fused into a single WMMA kernel per layer (fp32 16x16x4 WMMA).
// ---------------------------------------------------------------------------

typedef float v2f __attribute__((ext_vector_type(2)));
typedef float v8f __attribute__((ext_vector_type(8)));

#define NFEAT 64
#define WPITCH 66   // LDS row pitch in floats: bank = (66n + k) % 64 = (2n+k)%64 -> conflict-free for n in 0..15

// ---------------------------------------------------------------------------
__global__ void zero_f32(float* __restrict__ p, int n) {
    int i = blockIdx.x * blockDim.x + threadIdx.x;
    if (i < n) p[i] = 0.0f;
}

// degree count: cnt[dst] += 1
__global__ void degree_kernel(const int* __restrict__ dst, float* __restrict__ cnt, int n_edges) {
    int e = blockIdx.x * blockDim.x + threadIdx.x;
    if (e < n_edges) atomicAdd(&cnt[dst[e]], 1.0f);
}

// scatter-add: agg[dst, :] += feat[src, :]   (thread = one edge x 4 features)
__global__ void scatter_add_kernel(const float* __restrict__ feat,
                                   const int* __restrict__ src,
                                   const int* __restrict__ dst,
                                   float* __restrict__ agg,
                                   int n_edges) {
    long long idx = (long long)blockIdx.x * blockDim.x + threadIdx.x;
    int e = (int)(idx >> 4);
    if (e >= n_edges) return;
    int f = ((int)idx & 15) * 4;
    int s = src[e];
    int d = dst[e];
    const float4 v = *(const float4*)(feat + (long long)s * NFEAT + f);
    float* base = agg + (long long)d * NFEAT + f;
    atomicAdd(base + 0, v.x);
    atomicAdd(base + 1, v.y);
    atomicAdd(base + 2, v.z);
    atomicAdd(base + 3, v.w);
}

// ---------------------------------------------------------------------------
// Fused SAGE linear:  out[n,:] = act( (agg[n,:]/max(cnt,1)) @ Wl^T + b + h[n,:] @ Wr^T )
// blockDim = 128 (4 waves). Each wave owns 16 nodes and produces all 64 outputs
// via chained V_WMMA_F32_16X16X4_F32 (16 K-steps x 4 N-tiles x 2 weight mats).
// Weights staged in LDS with padded pitch; EXEC all-ones (loads clamped, stores
// predicated) per WMMA restrictions.
// ---------------------------------------------------------------------------
__global__ void sage_linear_wmma(const float* __restrict__ agg,
                                 const float* __restrict__ h,
                                 const float* __restrict__ cnt,
                                 const float* __restrict__ Wl,   // [64,64] row-major (out,in)
                                 const float* __restrict__ Wr,   // [64,64]
                                 const float* __restrict__ bias, // [64]
                                 float* __restrict__ out,
                                 int n_nodes, int do_relu) {
    __shared__ float sWl[NFEAT * WPITCH];
    __shared__ float sWr[NFEAT * WPITCH];

    for (int i = threadIdx.x; i < NFEAT * NFEAT; i += blockDim.x) {
        int n = i >> 6, k = i & 63;
        sWl[n * WPITCH + k] = Wl[i];
        sWr[n * WPITCH + k] = Wr[i];
    }
    __syncthreads();

    const int wave = threadIdx.x >> 5;
    const int lane = threadIdx.x & 31;
    const int row  = lane & 15;      // A-matrix M within 16-row tile
    const int half = lane >> 4;      // K sub-pair select (0 -> K%4 in {0,1}, 1 -> {2,3})
    const int kb   = half * 2;
    const int ncol = lane & 15;      // B/C/D column within 16-col tile

    const int node_base = (blockIdx.x * 4 + wave) * 16;
    int node = node_base + row;
    int nclamp = node < n_nodes ? node : (n_nodes - 1);  // keep EXEC all-ones

    float deg = cnt[nclamp];
    float inv = 1.0f / fmaxf(deg, 1.0f);

    // Preload both A operands for all 16 K-steps (v2f each: 64 VGPRs total).
    const float* arow = agg + (long long)nclamp * NFEAT;
    const float* hrow = h   + (long long)nclamp * NFEAT;
    v2f a_agg[16], a_h[16];
#pragma unroll
    for (int s = 0; s < 16; ++s) {
        int k = 4 * s + kb;                       // even -> 8B aligned
        v2f av = *(const v2f*)(arow + k);
        a_agg[s].x = av.x * inv;
        a_agg[s].y = av.y * inv;
        a_h[s] = *(const v2f*)(hrow + k);
    }

#pragma unroll
    for (int t = 0; t < 4; ++t) {                 // 4 N-tiles of 16 output cols
        const int n = t * 16 + ncol;
        const float bv = bias[n];
        v8f c;
#pragma unroll
        for (int r = 0; r < 8; ++r) c[r] = bv;    // bias baked into accumulator

        const float* wl = &sWl[n * WPITCH];
        const float* wr = &sWr[n * WPITCH];
#pragma unroll
        for (int s = 0; s < 16; ++s) {            // agg @ Wl^T
            v2f bf = *(const v2f*)(wl + 4 * s + kb);
            c = __builtin_amdgcn_wmma_f32_16x16x4_f32(
                    false, a_agg[s], false, bf, (short)0, c, false, false);
        }
#pragma unroll
        for (int s = 0; s < 16; ++s) {            // + h @ Wr^T
            v2f bf = *(const v2f*)(wr + 4 * s + kb);
            c = __builtin_amdgcn_wmma_f32_16x16x4_f32(
                    false, a_h[s], false, bf, (short)0, c, false, false);
        }

        // D layout: VGPR r -> (M = r + 8*half, N = ncol)
#pragma unroll
        for (int r = 0; r < 8; ++r) {
            int m = r + half * 8;
            int onode = node_base + m;
            if (onode < n_nodes) {
                float v = c[r];
                if (do_relu) v = fmaxf(v, 0.0f);
                out[(long long)onode * NFEAT + n] = v;
            }
        }
    }
}

// ---------------------------------------------------------------------------
extern "C" void kernel_launch(void* const* d_in, const int* in_sizes, int n_in,
                              void* d_out, int out_size, void* d_ws, size_t ws_size,
                              hipStream_t stream) {
    const float* x   = (const float*)d_in[0];   // [N,64]
    const int*   ei  = (const int*)  d_in[1];   // [2,E]
    const float* W1l = (const float*)d_in[2];
    const float* W1r = (const float*)d_in[3];
    const float* b1  = (const float*)d_in[4];
    const float* W2l = (const float*)d_in[5];
    const float* W2r = (const float*)d_in[6];
    const float* b2  = (const float*)d_in[7];
    float* out = (float*)d_out;

    const int N = in_sizes[0] / NFEAT;          // 100000
    const int E = in_sizes[1] / 2;              // 1250000
    const int* src = ei;
    const int* dst = ei + E;

    // workspace layout (floats): cnt[N] | agg[N*64] | h1[N*64]
    float* cnt = (float*)d_ws;
    float* agg = cnt + N;
    float* h1  = agg + (long long)N * NFEAT;

    const int ZT = 256;
    const int zn1 = N * (NFEAT + 1);            // cnt + agg contiguous
    hipLaunchKernelGGL(zero_f32, dim3((zn1 + ZT - 1) / ZT), dim3(ZT), 0, stream, cnt, zn1);

    hipLaunchKernelGGL(degree_kernel, dim3((E + ZT - 1) / ZT), dim3(ZT), 0, stream, dst, cnt, E);

    const long long sc_threads = (long long)E * 16;
    const int sc_blocks = (int)((sc_threads + ZT - 1) / ZT);
    hipLaunchKernelGGL(scatter_add_kernel, dim3(sc_blocks), dim3(ZT), 0, stream, x, src, dst, agg, E);

    const int lin_blocks = (N + 63) / 64;       // 64 nodes per block (4 waves x 16)
    hipLaunchKernelGGL(sage_linear_wmma, dim3(lin_blocks), dim3(128), 0, stream,
                       agg, x, cnt, W1l, W1r, b1, h1, N, /*relu=*/1);

    const int zn2 = N * NFEAT;
    hipLaunchKernelGGL(zero_f32, dim3((zn2 + ZT - 1) / ZT), dim3(ZT), 0, stream, agg, zn2);

    hipLaunchKernelGGL(scatter_add_kernel, dim3(sc_blocks), dim3(ZT), 0, stream, h1, src, dst, agg, E);

    hipLaunchKernelGGL(sage_linear_wmma, dim3(lin_blocks), dim3(128), 0, stream,
                       agg, h1, cnt, W2l, W2r, b2, out, N, /*relu=*/0);
}